// GCNAutoencoder_89610197664064
// MI455X (gfx1250) — compile-verified
//
#include <hip/hip_runtime.h>
#include <math.h>

// Problem constants (match reference)
#define T_STEPS 12
#define NNODES  10000
#define NEDGES  320000
#define IN_DIM  64
#define HID_DIM 128
#define EMB_DIM 64
#define G3      (3 * EMB_DIM)   // 192

typedef __attribute__((ext_vector_type(2))) float v2f;
typedef __attribute__((ext_vector_type(8))) float v8f;

// ---------------------------------------------------------------- elementwise

__global__ void k_zero(float* __restrict__ p, int n) {
    int i = blockIdx.x * blockDim.x + threadIdx.x;
    if (i < n) p[i] = 0.0f;
}

__global__ void k_copy(const float* __restrict__ src, float* __restrict__ dst, int n) {
    int i = blockIdx.x * blockDim.x + threadIdx.x;
    if (i < n) dst[i] = src[i];
}

// W[rows][cols] -> Wt[cols][rows]
__global__ void k_transpose(const float* __restrict__ W, float* __restrict__ Wt,
                            int rows, int cols) {
    int i = blockIdx.x * blockDim.x + threadIdx.x;
    if (i < rows * cols) {
        int r = i / cols, c = i % cols;
        Wt[c * rows + r] = W[i];
    }
}

// deg = segment_sum(ew, col) over E real edges + N self-loops (weight 1)
__global__ void k_deg(const int* __restrict__ ei, const float* __restrict__ ew,
                      float* __restrict__ deg) {
    int e = blockIdx.x * blockDim.x + threadIdx.x;
    if (e < NEDGES) {
        atomicAdd(&deg[ei[NEDGES + e]], ew[e]);
    } else if (e < NEDGES + NNODES) {
        atomicAdd(&deg[e - NEDGES], 1.0f);
    }
}

__global__ void k_dinv(const float* __restrict__ deg, float* __restrict__ dinv) {
    int i = blockIdx.x * blockDim.x + threadIdx.x;
    if (i < NNODES) {
        float d = deg[i];
        dinv[i] = (d > 0.0f) ? rsqrtf(d) : 0.0f;
    }
}

__global__ void k_norm(const int* __restrict__ ei, const float* __restrict__ ew,
                       const float* __restrict__ dinv, float* __restrict__ nrm) {
    int e = blockIdx.x * blockDim.x + threadIdx.x;
    if (e < NEDGES) {
        nrm[e] = dinv[ei[e]] * ew[e] * dinv[ei[NEDGES + e]];
    } else if (e < NEDGES + NNODES) {
        float d = dinv[e - NEDGES];
        nrm[e] = d * d;
    }
}

// agg[col] += h[row] * norm  over (E + N self-loop) edges, F features each
__global__ void k_scatter(const int* __restrict__ ei, const float* __restrict__ nrm,
                          const float* __restrict__ h, float* __restrict__ agg, int F) {
    int i = blockIdx.x * blockDim.x + threadIdx.x;
    int total = (NEDGES + NNODES) * F;
    if (i >= total) return;
    int e = i / F;
    int f = i - e * F;
    int r = (e < NEDGES) ? ei[e] : (e - NEDGES);
    int c = (e < NEDGES) ? ei[NEDGES + e] : (e - NEDGES);
    atomicAdd(&agg[c * F + f], h[r * F + f] * nrm[e]);
}

__global__ void k_bias_act(float* __restrict__ h, const float* __restrict__ b,
                           int n, int F, int relu) {
    int i = blockIdx.x * blockDim.x + threadIdx.x;
    if (i >= n * F) return;
    float v = h[i] + b[i % F];
    if (relu) v = fmaxf(v, 0.0f);
    h[i] = v;
}

// GRU gates: gate order r, z, n (torch layout); h updated in place
__global__ void k_gru_gate(const float* __restrict__ gi, const float* __restrict__ gh,
                           float* __restrict__ h) {
    int i = blockIdx.x * blockDim.x + threadIdx.x;
    if (i >= NNODES * EMB_DIM) return;
    int n = i / EMB_DIM, j = i - n * EMB_DIM;
    const float* gin = gi + n * G3;
    const float* ghn = gh + n * G3;
    float r  = 1.0f / (1.0f + expf(-(gin[j] + ghn[j])));
    float zg = 1.0f / (1.0f + expf(-(gin[EMB_DIM + j] + ghn[EMB_DIM + j])));
    float ng = tanhf(gin[2 * EMB_DIM + j] + r * ghn[2 * EMB_DIM + j]);
    h[i] = (1.0f - zg) * ng + zg * h[i];
}

// ---------------------------------------------------------------- WMMA GEMMs
// Native fp32 WMMA: V_WMMA_F32_16X16X4_F32 (D = A(16x4) * B(4x16) + C).
// fp32 A 16x4 tile: lane l(0..15) holds A[l][2h], A[l][2h+1]  (h = lane>=16)
// fp32 B 4x16 tile: lane l(0..15) holds B[2h][l], B[2h+1][l]
// fp32 D 16x16:     vgpr r -> row r + 8h, col lane&15

// C[M,Nc] = A[M,K] @ B[K,Nc] (+bias broadcast over rows), row-major.
// Block = (16-col strip, 8-tile row group). B panel staged TRANSPOSED in LDS so
// each lane's (B[ka][c], B[ka+1][c]) pair is one ds_load_b64; A pairs are
// contiguous global_load_b64. All operands preloaded -> K/4 back-to-back WMMAs.
template <int K>
__global__ void __launch_bounds__(256)
k_gemm_wmma_f32(const float* __restrict__ A, const float* __restrict__ B,
                const float* __restrict__ bias, float* __restrict__ C,
                int M, int Nc) {
    constexpr int S = K / 4;
    __shared__ float BsT[16 * K];            // [col][k], <= 8 KB
    const int tj0 = blockIdx.x * 16;
    const int tid = threadIdx.x;

    for (int idx = tid; idx < 16 * K; idx += 256) {
        int k = idx >> 4;                    // coalesced over 16 columns per row
        int c = idx & 15;
        BsT[c * K + k] = B[k * Nc + tj0 + c];
    }
    __syncthreads();

    int w  = tid >> 5;
    int ti = blockIdx.y * 8 + w;
    if (ti >= (M >> 4)) return;              // wave-uniform: EXEC stays full

    int lane = tid & 31;
    int half = lane >> 4;
    int l16  = lane & 15;

    const float* Ar = A + (size_t)(ti * 16 + l16) * K;
    const float* Br = BsT + l16 * K;

    v2f areg[S], breg[S];
    #pragma unroll
    for (int s = 0; s < S; ++s) {
        int ka = 4 * s + 2 * half;
        areg[s] = *(const v2f*)(Ar + ka);    // global_load_b64
        breg[s] = *(const v2f*)(Br + ka);    // ds_load_b64
    }

    v8f acc = {};
    #pragma unroll
    for (int s = 0; s < S; ++s)
        acc = __builtin_amdgcn_wmma_f32_16x16x4_f32(
                  false, areg[s], false, breg[s], (short)0, acc, false, false);

    float bv = bias ? bias[tj0 + l16] : 0.0f;
    float* Cb = C + (size_t)(ti * 16) * Nc + tj0;
    #pragma unroll
    for (int r = 0; r < 8; ++r) {
        int m = r + 8 * half;
        Cb[m * Nc + l16] = acc[r] + bv;
    }
}

// od = softplus(z @ z^T + dec_bias), z:[N,EMB].
// Block = 8 waves on a 2(ti) x 4(tj) tile group; 96 z-rows (24 KB) staged in LDS
// with coalesced float4 loads, then ds_load_b64 pairs feed 16 straight WMMAs.
__global__ void __launch_bounds__(256)
k_decoder_wmma(const float* __restrict__ z, const float* __restrict__ dec_bias,
               float* __restrict__ out) {
    __shared__ float As[32 * EMB_DIM];       // 8 KB
    __shared__ float Bs[64 * EMB_DIM];       // 16 KB
    const int nt   = NNODES / 16;            // 625
    const int ntjg = (nt + 3) / 4;           // 157

    int g   = blockIdx.x;
    int gi_ = g / ntjg;
    int gj  = g - gi_ * ntjg;
    int arow0 = gi_ * 32;
    int brow0 = gj * 64;

    int tid = threadIdx.x;
    for (int idx = tid; idx < 1536; idx += 256) {   // 96 rows * 16 float4
        int row = idx >> 4;
        int c4  = (idx & 15) << 2;
        int grow = (row < 32) ? (arow0 + row) : (brow0 + row - 32);
        float4 v = make_float4(0.0f, 0.0f, 0.0f, 0.0f);
        if (grow < NNODES) v = *(const float4*)(z + (size_t)grow * EMB_DIM + c4);
        float* dst = (row < 32) ? (As + row * EMB_DIM + c4)
                                : (Bs + (row - 32) * EMB_DIM + c4);
        *(float4*)dst = v;
    }
    __syncthreads();

    int w   = tid >> 5;
    int til = w >> 2;
    int tjl = w & 3;
    int ti  = gi_ * 2 + til;
    int tj  = gj * 4 + tjl;
    if (ti >= nt || tj >= nt) return;        // wave-uniform edge skip

    int lane = tid & 31;
    int half = lane >> 4;
    int l16  = lane & 15;

    const float* Ar = As + (til * 16 + l16) * EMB_DIM;
    const float* Br = Bs + (tjl * 16 + l16) * EMB_DIM;

    v2f a[16], b[16];
    #pragma unroll
    for (int s = 0; s < 16; ++s) {
        int ka = 4 * s + 2 * half;
        a[s] = *(const v2f*)(Ar + ka);       // ds_load_b64
        b[s] = *(const v2f*)(Br + ka);       // ds_load_b64
    }

    v8f acc = {};
    #pragma unroll
    for (int s = 0; s < 16; ++s)
        acc = __builtin_amdgcn_wmma_f32_16x16x4_f32(
                  false, a[s], false, b[s], (short)0, acc, false, false);

    float db = dec_bias[0];
    long long base = (long long)(ti * 16) * NNODES + tj * 16;
    #pragma unroll
    for (int r = 0; r < 8; ++r) {
        int m = r + 8 * half;
        float v = acc[r] + db;
        float sp = (v > 20.0f) ? v : log1pf(expf(v));
        out[base + (long long)m * NNODES + l16] = sp;
    }
}

// ---------------------------------------------------------------- launch

static inline int cdiv(int a, int b) { return (a + b - 1) / b; }

extern "C" void kernel_launch(void* const* d_in, const int* in_sizes, int n_in,
                              void* d_out, int out_size, void* d_ws, size_t ws_size,
                              hipStream_t stream) {
    const float* x_seq    = (const float*)d_in[0];   // [T,N,IN]
    const int*   ei       = (const int*)  d_in[1];   // [2,E]
    const float* ew       = (const float*)d_in[2];   // [E]
    const float* W1       = (const float*)d_in[3];   // [IN,HID]
    const float* b1       = (const float*)d_in[4];   // [HID]
    const float* W2       = (const float*)d_in[5];   // [HID,EMB]
    const float* b2       = (const float*)d_in[6];   // [EMB]
    const float* W_ih     = (const float*)d_in[7];   // [3H,EMB] (torch layout)
    const float* W_hh     = (const float*)d_in[8];   // [3H,EMB]
    const float* b_ih     = (const float*)d_in[9];   // [3H]
    const float* b_hh     = (const float*)d_in[10];  // [3H]
    const float* dec_bias = (const float*)d_in[11];  // [1]
    float* out = (float*)d_out;

    // workspace carve-up (floats)
    float* ws   = (float*)d_ws;
    float* deg  = ws;  ws += NNODES;
    float* dinv = ws;  ws += NNODES;
    float* nrm  = ws;  ws += NEDGES + NNODES;
    float* WihT = ws;  ws += EMB_DIM * G3;           // [EMB,3H]
    float* WhhT = ws;  ws += EMB_DIM * G3;
    float* xw   = ws;  ws += (size_t)NNODES * HID_DIM;
    float* agg  = ws;  ws += (size_t)NNODES * HID_DIM;
    float* zseq = ws;  ws += (size_t)T_STEPS * NNODES * EMB_DIM;
    float* gi   = ws;  ws += (size_t)NNODES * G3;
    float* gh   = ws;  ws += (size_t)NNODES * G3;
    float* h    = ws;  ws += (size_t)NNODES * EMB_DIM;

    const int B = 256;
    const int MGRP = cdiv(NNODES / 16, 8);           // 79

    // --- graph normalization ---
    k_zero<<<cdiv(NNODES, B), B, 0, stream>>>(deg, NNODES);
    k_deg<<<cdiv(NEDGES + NNODES, B), B, 0, stream>>>(ei, ew, deg);
    k_dinv<<<cdiv(NNODES, B), B, 0, stream>>>(deg, dinv);
    k_norm<<<cdiv(NEDGES + NNODES, B), B, 0, stream>>>(ei, ew, dinv, nrm);

    // --- transpose GRU weights once: [3H,EMB] -> [EMB,3H] ---
    k_transpose<<<cdiv(G3 * EMB_DIM, B), B, 0, stream>>>(W_ih, WihT, G3, EMB_DIM);
    k_transpose<<<cdiv(G3 * EMB_DIM, B), B, 0, stream>>>(W_hh, WhhT, G3, EMB_DIM);

    // --- GCN encoder per timestep ---
    for (int t = 0; t < T_STEPS; ++t) {
        const float* xt = x_seq + (size_t)t * NNODES * IN_DIM;
        float* zt = zseq + (size_t)t * NNODES * EMB_DIM;

        k_gemm_wmma_f32<IN_DIM><<<dim3(HID_DIM / 16, MGRP), 256, 0, stream>>>(
            xt, W1, nullptr, xw, NNODES, HID_DIM);
        k_zero<<<cdiv(NNODES * HID_DIM, B), B, 0, stream>>>(agg, NNODES * HID_DIM);
        k_scatter<<<cdiv((NEDGES + NNODES) * HID_DIM, B), B, 0, stream>>>(ei, nrm, xw, agg, HID_DIM);
        k_bias_act<<<cdiv(NNODES * HID_DIM, B), B, 0, stream>>>(agg, b1, NNODES, HID_DIM, 1);

        k_gemm_wmma_f32<HID_DIM><<<dim3(EMB_DIM / 16, MGRP), 256, 0, stream>>>(
            agg, W2, nullptr, xw, NNODES, EMB_DIM);
        k_zero<<<cdiv(NNODES * EMB_DIM, B), B, 0, stream>>>(zt, NNODES * EMB_DIM);
        k_scatter<<<cdiv((NEDGES + NNODES) * EMB_DIM, B), B, 0, stream>>>(ei, nrm, xw, zt, EMB_DIM);
        k_bias_act<<<cdiv(NNODES * EMB_DIM, B), B, 0, stream>>>(zt, b2, NNODES, EMB_DIM, 0);
    }

    // --- GRU scan (bias fused into GEMM epilogue) ---
    k_zero<<<cdiv(NNODES * EMB_DIM, B), B, 0, stream>>>(h, NNODES * EMB_DIM);
    for (int t = 0; t < T_STEPS; ++t) {
        const float* zt = zseq + (size_t)t * NNODES * EMB_DIM;
        k_gemm_wmma_f32<EMB_DIM><<<dim3(G3 / 16, MGRP), 256, 0, stream>>>(
            zt, WihT, b_ih, gi, NNODES, G3);
        k_gemm_wmma_f32<EMB_DIM><<<dim3(G3 / 16, MGRP), 256, 0, stream>>>(
            h, WhhT, b_hh, gh, NNODES, G3);
        k_gru_gate<<<cdiv(NNODES * EMB_DIM, B), B, 0, stream>>>(gi, gh, h);
    }

    // --- decoder: softplus(z@z^T + bias) -> out[0 .. N*N) ---
    {
        const int nt = NNODES / 16;                  // 625
        int blocks = cdiv(nt, 2) * cdiv(nt, 4);      // 313 * 157
        k_decoder_wmma<<<blocks, 256, 0, stream>>>(h, dec_bias, out);
    }
    // z (final hidden) -> out tail
    k_copy<<<cdiv(NNODES * EMB_DIM, B), B, 0, stream>>>(h, out + (size_t)NNODES * NNODES,
                                                        NNODES * EMB_DIM);
}